// MaskedScatterOldDecomp_4269197492488
// MI455X (gfx1250) — compile-verified
//
#include <hip/hip_runtime.h>
#include <stdint.h>

// ---------------------------------------------------------------------------
// MaskedScatterOldDecomp on gfx1250 (MI455X):
//   The reference's full 33.5M-element cumsum collapses analytically to an
//   8192-element exclusive row-prefix:
//     srow[s] = mask[s] ? excl_prefix(mask)[s] : -1
//     out[s,:] = (srow[s] >= 0) ? source[srow[s],:] : inputs_embeds[s,:]
//   Total traffic: 2 x 128 MiB (read one row source + write), zero FLOPs
//   => HBM-bound, ~11.5 us at 23.3 TB/s.
//
// Phase 1: single-block scan of the 8192-entry mask.
// Phase 2: per-row 16 KB copy via CDNA5 async global<->LDS DMA
//          (global_load_async_to_lds_b128 / global_store_async_from_lds_b128,
//           ASYNCcnt). Data is touched exactly once, so both directions use
//          non-temporal hints to keep the 268 MB stream out of the 192 MB L2.
// ---------------------------------------------------------------------------

__device__ __forceinline__ uint64_t uniform_u64(uint64_t v) {
  uint32_t lo = (uint32_t)v;
  uint32_t hi = (uint32_t)(v >> 32);
  lo = (uint32_t)__builtin_amdgcn_readfirstlane((int)lo);
  hi = (uint32_t)__builtin_amdgcn_readfirstlane((int)hi);
  return ((uint64_t)hi << 32) | (uint64_t)lo;
}

// ---------------- Phase 1: exclusive prefix sum of the row mask -------------
__global__ __launch_bounds__(1024) void mask_scan_kernel(
    const int* __restrict__ mask, int* __restrict__ srow, int S) {
  __shared__ int part[1024];
  const int t = threadIdx.x;
  const int nt = 1024;
  const int epb = (S + nt - 1) / nt;  // 8 for S=8192
  const int b0 = t * epb;

  int sum = 0;
  if (b0 + epb <= S) {
#pragma unroll
    for (int i = 0; i < 8; ++i)
      if (i < epb) sum += (mask[b0 + i] != 0) ? 1 : 0;
    if (epb > 8)
      for (int i = 8; i < epb; ++i) sum += (mask[b0 + i] != 0) ? 1 : 0;
  } else {
    for (int i = b0; i < S; ++i) sum += (mask[i] != 0) ? 1 : 0;
  }
  part[t] = sum;
  __syncthreads();

  // Hillis-Steele inclusive scan over the 1024 per-thread partials.
  for (int off = 1; off < nt; off <<= 1) {
    int v = (t >= off) ? part[t - off] : 0;
    __syncthreads();
    part[t] += v;
    __syncthreads();
  }

  int run = part[t] - sum;  // exclusive prefix at start of this segment
  const int b1 = (b0 + epb <= S) ? (b0 + epb) : S;
  for (int i = b0; i < b1; ++i) {
    const int m = (mask[i] != 0) ? 1 : 0;
    srow[i] = m ? run : -1;
    run += m;
  }
}

// ---------------- Phase 2: row gather/copy via async LDS DMA ----------------
// One block per row. 256 threads * 4 float4 = 4096 floats = 16 KB row staged
// through LDS. Each wave's lanes read back only LDS bytes those same lanes
// wrote, so the wave-local s_wait_asynccnt 0 is the only sync needed — no
// barriers anywhere.
__global__ __launch_bounds__(256) void masked_row_copy_async(
    const float* __restrict__ inputs, const float* __restrict__ source,
    const int* __restrict__ srow, float* __restrict__ out, int D) {
  __shared__ __align__(16) float sbuf[4096];  // 16 KB staging buffer

  const int row = blockIdx.x;
  const int t = threadIdx.x;

  const int sr = srow[row];
  const float* srcp = (sr >= 0) ? (source + (size_t)sr * (size_t)D)
                                : (inputs + (size_t)row * (size_t)D);
  const uint64_t src = uniform_u64((uint64_t)(uintptr_t)srcp);
  const uint64_t dst =
      uniform_u64((uint64_t)(uintptr_t)(out + (size_t)row * (size_t)D));

  // Low 32 bits of the flat address of a __shared__ object == LDS byte offset.
  const unsigned ldsbase = (unsigned)(uintptr_t)(&sbuf[0]);

  const int nchunk = D >> 10;  // 1024 floats (=4 KB) per chunk; 4 for D=4096

  // Global -> LDS (async DMA, 512 B per wave per instruction), non-temporal:
  // every byte is consumed exactly once, keep it out of L2.
#pragma unroll 4
  for (int c = 0; c < nchunk; ++c) {
    const unsigned o = (unsigned)(t * 16 + c * 4096);  // byte offset in row
    const unsigned l = ldsbase + o;                    // matching LDS address
    asm volatile("global_load_async_to_lds_b128 %0, %1, %2 th:TH_LOAD_NT"
                 :
                 : "v"(l), "v"(o), "s"(src)
                 : "memory");
  }
  asm volatile("s_wait_asynccnt 0x0" ::: "memory");

  // LDS -> Global (async DMA), non-temporal store: output is never re-read.
#pragma unroll 4
  for (int c = 0; c < nchunk; ++c) {
    const unsigned o = (unsigned)(t * 16 + c * 4096);
    const unsigned l = ldsbase + o;
    asm volatile("global_store_async_from_lds_b128 %0, %1, %2 th:TH_STORE_NT"
                 :
                 : "v"(o), "v"(l), "s"(dst)
                 : "memory");
  }
  asm volatile("s_wait_asynccnt 0x0" ::: "memory");
}

// ---------------------------------------------------------------------------
extern "C" void kernel_launch(void* const* d_in, const int* in_sizes, int n_in,
                              void* d_out, int out_size, void* d_ws,
                              size_t ws_size, hipStream_t stream) {
  const float* inputs = (const float*)d_in[0];  // [S, D] f32
  const int* mask = (const int*)d_in[1];        // [S] int (0/1)
  const float* source = (const float*)d_in[2];  // [S, D] f32
  float* out = (float*)d_out;                   // [S, D] f32

  const int S = in_sizes[1];
  const int D = (S > 0) ? (in_sizes[0] / S) : 0;

  int* srow = (int*)d_ws;  // [S] ints of scratch

  mask_scan_kernel<<<1, 1024, 0, stream>>>(mask, srow, S);
  masked_row_copy_async<<<S, 256, 0, stream>>>(inputs, source, srow, out, D);
}